// SinkAttentionWrapper_16106127360376
// MI455X (gfx1250) — compile-verified
//
#include <hip/hip_runtime.h>
#include <hip/hip_bf16.h>
#include <math.h>

// ---------------- problem constants ----------------
#define HEADS   32
#define HD      128
#define DMODEL  4096          // HEADS*HD
#define TCACHE  8192          // MAX_CACHE
#define SINKSZ  4
#define CHUNK   512
#define NCH     (TCACHE / CHUNK)   // 16
#define INV_SQRT_D 0.08838834764831845f

// ---------------- workspace layout (floats) ----------------
#define WS_Q    0                       // 4096   q        (h*128+d)
#define WS_KN   4096                    // 4096   k_new
#define WS_VN   8192                    // 4096   v_new
#define WS_CTX  12288                   // 4096   attention context
#define WS_PMAX 16384                   // 32*16  per-(head,chunk) max
#define WS_PSUM 16896                   // 32*16  per-(head,chunk) expsum
#define WS_PCTX 17408                   // 32*16*128 partial context

typedef float v2f __attribute__((ext_vector_type(2)));
typedef float v8f __attribute__((ext_vector_type(8)));

// ---------------------------------------------------------------------------
// WMMA GEMV tile: out[col0..col0+15] = x(1x4096) * W(4096x4096) tile.
// 256 threads = 8 waves; wave w reduces K-range [512w, 512w+512) with
// V_WMMA_F32_16X16X4_F32 (A = x replicated over 16 rows -> all D rows equal).
// ---------------------------------------------------------------------------
__device__ __forceinline__ void gemv_tile_wmma(const float* __restrict__ xv,
                                               const float* __restrict__ W,
                                               float* __restrict__ out,
                                               int col0) {
  __shared__ __attribute__((aligned(16))) float xs[DMODEL];
  __shared__ float red[8 * 16];
  const int tid = threadIdx.x;
  for (int i = tid; i < DMODEL; i += 256) xs[i] = xv[i];
  __syncthreads();

  const int wave = tid >> 5;
  const int lane = tid & 31;
  const int half = lane >> 4;     // 0: K={0,1}, 1: K={2,3} of each 4-step
  const int n    = lane & 15;     // output column within tile

  v8f c = {};
  const int k0 = wave * 512;
  for (int kk = k0; kk < k0 + 512; kk += 4) {
    const int kb = kk + 2 * half;
    v2f a; a.x = xs[kb]; a.y = xs[kb + 1];
    const float* wp = W + (size_t)kb * DMODEL + col0 + n;
    v2f b; b.x = wp[0]; b.y = wp[DMODEL];
    c = __builtin_amdgcn_wmma_f32_16x16x4_f32(false, a, false, b,
                                              (short)0, c, false, false);
  }
  // row 0 of D: lanes 0..15, VGPR0 (c[0])
  if (lane < 16) red[wave * 16 + n] = c[0];
  __syncthreads();
  if (tid < 16) {
    float s = 0.f;
    for (int w = 0; w < 8; ++w) s += red[w * 16 + tid];
    out[col0 + tid] = s;
  }
}

// 768 tiles: matrix = tile>>8 (0:Wq 1:Wk 2:Wv), 16 columns each
__global__ void qkv_proj_kernel(const float* __restrict__ x,
                                const float* __restrict__ Wq,
                                const float* __restrict__ Wk,
                                const float* __restrict__ Wv,
                                float* __restrict__ ws) {
  const int tile = blockIdx.x;
  const int m    = tile >> 8;
  const int col0 = (tile & 255) << 4;
  const float* W = (m == 0) ? Wq : ((m == 1) ? Wk : Wv);
  gemv_tile_wmma(x, W, ws + m * DMODEL, col0);
}

__global__ void out_proj_kernel(const float* __restrict__ ws,
                                const float* __restrict__ Wo,
                                float* __restrict__ out) {
  gemv_tile_wmma(ws + WS_CTX, Wo, out, blockIdx.x << 4);
}

// ---------------------------------------------------------------------------
// Flash-decoding partial: block = (head, chunk of 512 keys), 256 threads.
// Phase 1: wave-per-key QK dots (coalesced 512B/row). Phase 2: chunk softmax.
// Phase 3: partial context accumulation over V.
// ---------------------------------------------------------------------------
__global__ void attn_partial_kernel(const float* __restrict__ Kc,
                                    const float* __restrict__ Vc,
                                    float* __restrict__ ws) {
  __shared__ __attribute__((aligned(16))) float qs[HD];
  __shared__ float sc[CHUNK];
  __shared__ float red[256];
  const int h     = blockIdx.x >> 4;
  const int chunk = blockIdx.x & 15;
  const int tid   = threadIdx.x;
  if (tid < HD) qs[tid] = ws[WS_Q + h * HD + tid];
  __syncthreads();

  const int wave = tid >> 5;
  const int lane = tid & 31;
  const float4 qv = ((const float4*)qs)[lane];

  // phase 1: scores
  const float* Kh = Kc + ((size_t)h * TCACHE + (size_t)chunk * CHUNK) * HD;
  for (int i = wave; i < CHUNK; i += 8) {
    const float4 kv = ((const float4*)(Kh + (size_t)i * HD))[lane];
    float p = qv.x * kv.x + qv.y * kv.y + qv.z * kv.z + qv.w * kv.w;
    for (int off = 16; off >= 1; off >>= 1) p += __shfl_xor(p, off, 32);
    if (lane == 0) sc[i] = p * INV_SQRT_D;
  }
  __syncthreads();

  // phase 2: chunk max, exp, sum
  red[tid] = fmaxf(sc[tid], sc[tid + 256]);
  __syncthreads();
  for (int off = 128; off >= 1; off >>= 1) {
    if (tid < off) red[tid] = fmaxf(red[tid], red[tid + off]);
    __syncthreads();
  }
  const float m = red[0];
  __syncthreads();
  const float e0 = __expf(sc[tid] - m);
  const float e1 = __expf(sc[tid + 256] - m);
  sc[tid] = e0; sc[tid + 256] = e1;
  red[tid] = e0 + e1;
  __syncthreads();
  for (int off = 128; off >= 1; off >>= 1) {
    if (tid < off) red[tid] += red[tid + off];
    __syncthreads();
  }
  if (tid == 0) {
    ws[WS_PMAX + h * NCH + chunk] = m;
    ws[WS_PSUM + h * NCH + chunk] = red[0];
  }

  // phase 3: partial context (128 dims x 2 key-segments)
  const int d   = tid & 127;
  const int seg = tid >> 7;
  const float* Vh = Vc + ((size_t)h * TCACHE + (size_t)chunk * CHUNK) * HD;
  float acc = 0.f;
  for (int t = seg * 256; t < seg * 256 + 256; ++t)
    acc += sc[t] * Vh[(size_t)t * HD + d];
  red[tid] = acc;
  __syncthreads();
  if (tid < 128)
    ws[WS_PCTX + ((size_t)(h * NCH + chunk)) * HD + tid] = red[tid] + red[tid + 128];
}

// ---------------------------------------------------------------------------
// Combine partials + new-token K/V contribution -> ctx. 32 blocks x 128 thr.
// ---------------------------------------------------------------------------
__global__ void attn_combine_kernel(float* __restrict__ ws) {
  __shared__ float red[128];
  __shared__ float bc[2];
  const int h = blockIdx.x;
  const int tid = threadIdx.x;

  red[tid] = ws[WS_Q + h * HD + tid] * ws[WS_KN + h * HD + tid];
  __syncthreads();
  for (int off = 64; off >= 1; off >>= 1) {
    if (tid < off) red[tid] += red[tid + off];
    __syncthreads();
  }
  const float s_new = red[0] * INV_SQRT_D;

  if (tid == 0) {
    float M = s_new;
    for (int i = 0; i < NCH; ++i) M = fmaxf(M, ws[WS_PMAX + h * NCH + i]);
    float den = __expf(s_new - M);
    for (int i = 0; i < NCH; ++i)
      den += ws[WS_PSUM + h * NCH + i] * __expf(ws[WS_PMAX + h * NCH + i] - M);
    bc[0] = M; bc[1] = den;
  }
  __syncthreads();
  const float M = bc[0], den = bc[1];

  float acc = __expf(s_new - M) * ws[WS_VN + h * HD + tid];
  for (int i = 0; i < NCH; ++i)
    acc += ws[WS_PCTX + ((size_t)(h * NCH + i)) * HD + tid] *
           __expf(ws[WS_PMAX + h * NCH + i] - M);
  ws[WS_CTX + h * HD + tid] = acc / den;
}

// ---------------------------------------------------------------------------
// Sink-preserving cache shift: [sink | recent[1:] | new], K and V fused.
// float4 lanes; one wave streams a full 512B row per iteration.
// ---------------------------------------------------------------------------
__global__ void cache_shift_kernel(const float4* __restrict__ Kc,
                                   const float4* __restrict__ Vc,
                                   const float* __restrict__ ws,
                                   float4* __restrict__ outK,
                                   float4* __restrict__ outV) {
  const size_t per = (size_t)HEADS * TCACHE * (HD / 4);   // 8,388,608 float4
  size_t i = (size_t)blockIdx.x * blockDim.x + threadIdx.x;
  const int isV = (i >= per);
  const size_t r = isV ? i - per : i;
  const float4* src = isV ? Vc : Kc;
  const float4* nw  = (const float4*)(ws + (isV ? WS_VN : WS_KN));
  float4* dst       = isV ? outV : outK;

  const size_t hp = r >> 5;          // h*8192 + p
  const int    j  = (int)(r & 31);
  const size_t h  = hp >> 13;
  const size_t p  = hp & (TCACHE - 1);

  float4 val;
  if (p < SINKSZ)            val = src[r];        // sink tokens kept
  else if (p < TCACHE - 1)   val = src[r + 32];   // shift window left by 1 row
  else                       val = nw[h * 32 + j]; // append new token
  dst[r] = val;
}

// ---------------------------------------------------------------------------
extern "C" void kernel_launch(void* const* d_in, const int* in_sizes, int n_in,
                              void* d_out, int out_size, void* d_ws, size_t ws_size,
                              hipStream_t stream) {
  const float* x  = (const float*)d_in[0];
  const float* Kc = (const float*)d_in[1];
  const float* Vc = (const float*)d_in[2];
  const float* Wq = (const float*)d_in[3];
  const float* Wk = (const float*)d_in[4];
  const float* Wv = (const float*)d_in[5];
  const float* Wo = (const float*)d_in[6];
  float* ws  = (float*)d_ws;
  float* out = (float*)d_out;

  float* outK = out + DMODEL;
  float* outV = outK + (size_t)HEADS * TCACHE * HD;

  // 1) q,k_new,v_new projections (WMMA f32 GEMV)
  qkv_proj_kernel<<<3 * 256, 256, 0, stream>>>(x, Wq, Wk, Wv, ws);
  // 2) split-K attention partials over the cached K/V
  attn_partial_kernel<<<HEADS * NCH, 256, 0, stream>>>(Kc, Vc, ws);
  // 3) combine partials + new token -> ctx
  attn_combine_kernel<<<HEADS, 128, 0, stream>>>(ws);
  // 4) output projection (WMMA f32 GEMV)
  out_proj_kernel<<<256, 256, 0, stream>>>(ws, Wo, out);
  // 5) fused sink-preserving KV cache shift (dominant bandwidth term)
  const size_t total4 = 2ull * HEADS * TCACHE * (HD / 4); // 16,777,216
  cache_shift_kernel<<<(unsigned)(total4 / 256), 256, 0, stream>>>(
      (const float4*)Kc, (const float4*)Vc, ws, (float4*)outK, (float4*)outV);
}